// ChemicalConstraints_56994216018243
// MI455X (gfx1250) — compile-verified
//
#include <hip/hip_runtime.h>
#include <math.h>

typedef float v2f __attribute__((ext_vector_type(2)));
typedef float v8f __attribute__((ext_vector_type(8)));

// ---------- chemistry tables ----------
__device__ __forceinline__ float dev_max_val(int z) {
  switch (z) {
    case 1:  return 1.f; case 6:  return 4.f; case 7:  return 3.f; case 8:  return 2.f;
    case 9:  return 1.f; case 15: return 5.f; case 16: return 6.f; case 17: return 1.f;
    case 35: return 1.f; case 53: return 1.f; default: return 4.f;
  }
}
__device__ __forceinline__ float dev_vdw(int z) {
  switch (z) {
    case 1:  return 1.2f;  case 6:  return 1.7f;  case 7:  return 1.55f; case 8:  return 1.52f;
    case 9:  return 1.47f; case 15: return 1.8f;  case 16: return 1.8f;  case 17: return 1.75f;
    case 35: return 1.85f; case 53: return 1.98f; default: return 1.6f;
  }
}
__device__ __forceinline__ float dev_bond(int a, int b) {
  if (a > b) { int t = a; a = b; b = t; }
  switch (a * 64 + b) {
    case 6*64+6:   return 1.54f;
    case 6*64+7:   return 1.47f;
    case 6*64+8:   return 1.43f;
    case 6*64+16:  return 1.82f;
    case 6*64+9:   return 1.35f;
    case 6*64+17:  return 1.77f;
    case 1*64+6:   return 1.09f;
    case 7*64+7:   return 1.45f;
    case 7*64+8:   return 1.40f;
    case 1*64+7:   return 1.01f;
    case 8*64+8:   return 1.48f;
    case 1*64+8:   return 0.96f;
    case 16*64+16: return 2.05f;
    case 8*64+15:  return 1.63f;
    default:       return 1.5f;
  }
}

// ---------- phase 0: zero accumulators ----------
__global__ void k_zero(float* __restrict__ counts, float4* __restrict__ delta,
                       float* __restrict__ lossAcc, int N) {
  int i = blockIdx.x * blockDim.x + threadIdx.x;
  if (i < N) {
    counts[i] = 0.f;
    delta[i] = make_float4(0.f, 0.f, 0.f, 0.f);
  }
  if (i < 4) lossAcc[i] = 0.f;
}

// ---------- phase 1a: out-degree scatter-add ----------
__global__ void k_counts(const int* __restrict__ edges, float* __restrict__ counts, int E) {
  int e = blockIdx.x * blockDim.x + threadIdx.x;
  if (e < E) atomicAdd(&counts[edges[e]], 1.f);
}

// ---------- phase 1b: violation + loss1 + init padded pos1 ----------
__global__ void k_atoms(const float* __restrict__ pos, const int* __restrict__ at,
                        const float* __restrict__ counts, float* __restrict__ violation,
                        float4* __restrict__ pos1, float* __restrict__ lossAcc, int N) {
  int i = blockIdx.x * blockDim.x + threadIdx.x;
  if (i >= N) return;
  float v = fmaxf(counts[i] - dev_max_val(at[i]), 0.f);
  violation[i] = v;
  if (v > 0.f) atomicAdd(lossAcc + 0, v * v);
  pos1[i] = make_float4(pos[3*i], pos[3*i+1], pos[3*i+2], 0.f);
}

// ---------- phase 1c: valence push (parallelized scan) ----------
__global__ void k_push(const float* __restrict__ pos, const int* __restrict__ edges,
                       const float* __restrict__ violation, float4* __restrict__ pos1, int E) {
  int e = blockIdx.x * blockDim.x + threadIdx.x;
  if (e >= E) return;
  int r = edges[e], c = edges[E + e];
  float v = violation[r];
  if (v <= 0.f) return;
  float dx = pos[3*r+0] - pos[3*c+0];
  float dy = pos[3*r+1] - pos[3*c+1];
  float dz = pos[3*r+2] - pos[3*c+2];
  float dist = sqrtf(dx*dx + dy*dy + dz*dz) + 1e-8f;
  float s = v * 1e-3f / dist;
  float* pr = (float*)(pos1 + r);
  atomicAdd(pr + 0, dx * s);
  atomicAdd(pr + 1, dy * s);
  atomicAdd(pr + 2, dz * s);
}

// ---------- phase 2: bond-length correction ----------
__global__ void k_bond(const float4* __restrict__ pos1, const int* __restrict__ edges,
                       const int* __restrict__ at, float4* __restrict__ delta,
                       float* __restrict__ lossAcc, int E) {
  int e = blockIdx.x * blockDim.x + threadIdx.x;
  if (e >= E) return;
  int r = edges[e], c = edges[E + e];
  float4 pr = pos1[r], pc = pos1[c];
  float bx = pr.x - pc.x, by = pr.y - pc.y, bz = pr.z - pc.z;
  float cur = sqrtf(bx*bx + by*by + bz*bz);
  float tgt = dev_bond(at[r], at[c]);
  float diff = cur - tgt;
  atomicAdd(lossAcc + 1, diff * diff / (float)E);
  float ratio = fminf(fmaxf(tgt / (cur + 1e-8f), 0.98f), 1.02f);
  float s = (ratio - 1.f) * 0.01f * 0.5f;   // adj * 0.5
  float ax = bx * s, ay = by * s, az = bz * s;
  float* dr = (float*)(delta + r);
  float* dc = (float*)(delta + c);
  atomicAdd(dr + 0,  ax); atomicAdd(dr + 1,  ay); atomicAdd(dr + 2,  az);
  atomicAdd(dc + 0, -ax); atomicAdd(dc + 1, -ay); atomicAdd(dc + 2, -az);
}

// ---------- phase 2b: apply deltas -> pos2 (padded, w=0) + prescaled radii ----------
__global__ void k_apply(const float4* __restrict__ pos1, const float4* __restrict__ delta,
                        const int* __restrict__ at, float4* __restrict__ pos2,
                        float* __restrict__ rad08, int N) {
  int i = blockIdx.x * blockDim.x + threadIdx.x;
  if (i >= N) return;
  float4 p = pos1[i], d = delta[i];
  pos2[i] = make_float4(p.x + d.x, p.y + d.y, p.z + d.z, 0.f);
  rad08[i] = dev_vdw(at[i]) * 0.8f;   // min_d = (r_i + r_j)*0.8 = rad08_i + rad08_j
}

// ---------- phase 3: N x N steric via V_WMMA_F32_16X16X4_F32 ----------
// One wave owns a 16-row block; loops over all 16-column tiles.
// A: 16x4 (rows, xyz0), B: 4x16 (cols, xyz0), C: 16x16 f32 dot products.
__global__ __launch_bounds__(256, 1) void k_steric(
    const float4* __restrict__ pos2, const float* __restrict__ rad08,
    float* __restrict__ coeffsum, float4* __restrict__ cp,
    float* __restrict__ lossAcc, int N) {
  const int tid  = blockIdx.x * blockDim.x + threadIdx.x;
  const int lane = threadIdx.x & 31;
  const int wave = tid >> 5;
  const int half = lane >> 4;   // 0: K=0,1 ; 1: K=2,3
  const int l15  = lane & 15;
  const int rowBase = wave * 16;

  // A fragment: lane l holds components (2*half, 2*half+1) of row (rowBase + l15)
  const float2 af = *(reinterpret_cast<const float2*>(pos2 + rowBase + l15) + half);
  v2f a; a.x = af.x; a.y = af.y;

  // Per-lane row-side constants for the 8 C rows this lane touches (m = k + 8*half)
  float rrow[8], sqrow[8];
#pragma unroll
  for (int k = 0; k < 8; ++k) {
    int r = rowBase + 8 * half + k;
    float4 p = pos2[r];
    sqrow[k] = p.x*p.x + p.y*p.y + p.z*p.z;
    rrow[k]  = rad08[r];
  }

  float accC[8], accX[8], accY[8], accZ[8];
#pragma unroll
  for (int k = 0; k < 8; ++k) { accC[k] = accX[k] = accY[k] = accZ[k] = 0.f; }
  float lloss = 0.f;

  const int nTiles = N >> 4;
#pragma unroll 2
  for (int t = 0; t < nTiles; ++t) {
    const int n = t * 16 + l15;
    const float4 pn = pos2[n];
    const float rn  = rad08[n];
    // B fragment reuses pn: lane needs components (2*half, 2*half+1) of column n
    v2f b;
    b.x = half ? pn.z : pn.x;
    b.y = half ? pn.w : pn.y;   // pn.w == 0

    const float sqn = pn.x*pn.x + pn.y*pn.y + pn.z*pn.z;

    v8f c = {0.f, 0.f, 0.f, 0.f, 0.f, 0.f, 0.f, 0.f};
    c = __builtin_amdgcn_wmma_f32_16x16x4_f32(
        /*neg_a=*/false, a, /*neg_b=*/false, b,
        /*c_mod=*/(short)0, c, /*reuse_a=*/false, /*reuse_b=*/false);

#pragma unroll
    for (int k = 0; k < 8; ++k) {
      const int ri = rowBase + 8 * half + k;
      float d2c  = fmaxf(sqrow[k] + sqn - 2.f * c[k], 1e-20f);
      float rsq  = __builtin_amdgcn_rsqf(d2c);  // 1/dist
      float dist = d2c * rsq;                   // sqrt(d2c)
      float md   = rrow[k] + rn;
      float gap  = fmaxf(md - dist, 0.f);
      gap = (ri == n) ? 0.f : gap;              // diagonal: dist -> BIG in reference
      lloss = fmaf(gap, gap * 0.5f, lloss);
      // coeff: nonzero only when gap>0 (dist<md) and dist>1e-8; 0.005*0.5 = 0.0025
      float co = (dist > 1e-8f) ? gap * 0.0025f * rsq : 0.f;
      accC[k] += co;
      accX[k] = fmaf(co, pn.x, accX[k]);
      accY[k] = fmaf(co, pn.y, accY[k]);
      accZ[k] = fmaf(co, pn.z, accZ[k]);
    }
  }

  // reduce across the 16 lanes sharing each row (xor<16 stays within half)
#pragma unroll
  for (int off = 1; off < 16; off <<= 1) {
#pragma unroll
    for (int k = 0; k < 8; ++k) {
      accC[k] += __shfl_xor(accC[k], off, 32);
      accX[k] += __shfl_xor(accX[k], off, 32);
      accY[k] += __shfl_xor(accY[k], off, 32);
      accZ[k] += __shfl_xor(accZ[k], off, 32);
    }
  }
#pragma unroll
  for (int off = 1; off < 32; off <<= 1) lloss += __shfl_xor(lloss, off, 32);

  if (l15 == 0) {
#pragma unroll
    for (int k = 0; k < 8; ++k) {
      int r = rowBase + 8 * half + k;
      coeffsum[r] = accC[k];
      cp[r] = make_float4(accX[k], accY[k], accZ[k], 0.f);
    }
  }
  if (lane == 0) atomicAdd(lossAcc + 2, lloss);
}

// ---------- finalize: apply steric displacement + total loss ----------
__global__ void k_finalize(const float4* __restrict__ pos2, const float* __restrict__ coeffsum,
                           const float4* __restrict__ cp, const float* __restrict__ lossAcc,
                           float* __restrict__ out, int N) {
  int i = blockIdx.x * blockDim.x + threadIdx.x;
  if (i < N) {
    float4 p = pos2[i];
    float cs = coeffsum[i];
    float4 c = cp[i];
    out[3*i+0] = p.x + cs * p.x - c.x;
    out[3*i+1] = p.y + cs * p.y - c.y;
    out[3*i+2] = p.z + cs * p.z - c.z;
  }
  if (i == 0) out[3*N] = (lossAcc[0] + lossAcc[1] + lossAcc[2]) * 0.1f;
}

extern "C" void kernel_launch(void* const* d_in, const int* in_sizes, int n_in,
                              void* d_out, int out_size, void* d_ws, size_t ws_size,
                              hipStream_t stream) {
  const float* pos   = (const float*)d_in[0];
  const int*   edges = (const int*)d_in[1];   // [2, E] row-major: row = edges[e], col = edges[E+e]
  const int*   at    = (const int*)d_in[2];
  const int N = in_sizes[0] / 3;
  const int E = in_sizes[1] / 2;
  float* out = (float*)d_out;

  char* ws = (char*)d_ws;
  float4* pos1     = (float4*)ws; ws += (size_t)N * sizeof(float4);
  float4* delta    = (float4*)ws; ws += (size_t)N * sizeof(float4);
  float4* pos2     = (float4*)ws; ws += (size_t)N * sizeof(float4);
  float4* cp       = (float4*)ws; ws += (size_t)N * sizeof(float4);
  float*  counts   = (float*)ws;  ws += (size_t)N * sizeof(float);
  float*  violation= (float*)ws;  ws += (size_t)N * sizeof(float);
  float*  coeffsum = (float*)ws;  ws += (size_t)N * sizeof(float);
  float*  rad08    = (float*)ws;  ws += (size_t)N * sizeof(float);
  float*  lossAcc  = (float*)ws;

  const int bN = (N + 255) / 256;
  const int bE = (E + 255) / 256;

  k_zero   <<<bN, 256, 0, stream>>>(counts, delta, lossAcc, N);
  k_counts <<<bE, 256, 0, stream>>>(edges, counts, E);
  k_atoms  <<<bN, 256, 0, stream>>>(pos, at, counts, violation, pos1, lossAcc, N);
  k_push   <<<bE, 256, 0, stream>>>(pos, edges, violation, pos1, E);
  k_bond   <<<bE, 256, 0, stream>>>(pos1, edges, at, delta, lossAcc, E);
  k_apply  <<<bN, 256, 0, stream>>>(pos1, delta, at, pos2, rad08, N);

  // one wave per 16-row block; N=8192 -> 512 waves -> 64 blocks of 256 threads
  const int waves   = N / 16;
  const int threads = waves * 32;
  k_steric <<<threads / 256, 256, 0, stream>>>(pos2, rad08, coeffsum, cp, lossAcc, N);

  k_finalize<<<bN, 256, 0, stream>>>(pos2, coeffsum, cp, lossAcc, out, N);
}